// similar_attn_18700287607377
// MI455X (gfx1250) — compile-verified
//
#include <hip/hip_runtime.h>
#include <math.h>

// MI455X / gfx1250, wave32. All matmuls use V_WMMA_F32_16X16X4_F32 (fp32, K=4).
//
// Workspace layout (floats), total ~67.3 MB:
//   tq     : B*K*D   (trunc((tlt@WqT+bq)/sqrt(D)))
//   sk     : B*Q*D   (src@WkT+bk)
//   rowmax : B*K     (softmax-over-q row max, masked)
//   rowsum : B*K     (softmax-over-q row sum-exp, masked)

typedef __attribute__((ext_vector_type(2))) float v2f;
typedef __attribute__((ext_vector_type(4))) float v4f;
typedef __attribute__((ext_vector_type(8))) float v8f;
typedef __attribute__((ext_vector_type(4))) int v4i;

#define D_MODEL 512
#define BATCH 8
#define SEQ 2048
#define SCALE 22.62741699796952f   // sqrt(512)
#define SK_STRIDE 516              // 512 + 4 pad (bank-conflict-free, 64 banks)
#define ATT_STRIDE 132             // 128 + 4 pad

#define AS1 __attribute__((address_space(1)))
#define AS3 __attribute__((address_space(3)))

#if defined(__has_builtin)
#if __has_builtin(__builtin_amdgcn_global_load_async_to_lds_b128) && \
    __has_builtin(__builtin_amdgcn_s_wait_asynccnt)
#define USE_ASYNC_LDS 1
#endif
#endif
#ifndef USE_ASYNC_LDS
#define USE_ASYNC_LDS 0
#endif

__device__ __forceinline__ v8f wmma_f32_k4(v2f a, v2f b, v8f c) {
  // (neg_a, A, neg_b, B, c_mod, C, reuse_a, reuse_b) -> v_wmma_f32_16x16x4_f32
  return __builtin_amdgcn_wmma_f32_16x16x4_f32(false, a, false, b, (short)0, c,
                                               false, false);
}

__device__ __forceinline__ v8f zero8() {
  v8f z;
  #pragma unroll
  for (int i = 0; i < 8; ++i) z[i] = 0.0f;
  return z;
}

// Cooperative 16x512 tile copy global->LDS (row stride SK_STRIDE, padded).
// Uses gfx1250 GLOBAL_LOAD_ASYNC_TO_LDS_B128 (ASYNCcnt) when available.
__device__ __forceinline__ void tile_load_16x512(float* ldsBase,
                                                 const float* gBase, int tid) {
  for (int i = tid; i < 2048; i += 256) {
    const int g = i * 4, row = g >> 9, col = g & 511;
#if USE_ASYNC_LDS
    __builtin_amdgcn_global_load_async_to_lds_b128(
        (AS1 v4i*)(uintptr_t)(gBase + (size_t)row * D_MODEL + col),
        (AS3 v4i*)(uint32_t)(uintptr_t)&ldsBase[row * SK_STRIDE + col],
        /*offset=*/0, /*cpol=*/0);
#else
    *(v4f*)&ldsBase[row * SK_STRIDE + col] =
        *(const v4f*)(gBase + (size_t)row * D_MODEL + col);
#endif
  }
#if USE_ASYNC_LDS
  __builtin_amdgcn_s_wait_asynccnt(0);
#endif
}

// ---------------------------------------------------------------------------
// Out[m,n] = X[m,:] . W[n,:] + bias[n]   (optionally trunc(./SCALE))
// One block = one 16-row M tile x all 512 N columns. 8 waves x 4 N-tiles.
// ---------------------------------------------------------------------------
__global__ __launch_bounds__(256) void proj_kernel(
    const float* __restrict__ X, const float* __restrict__ W,
    const float* __restrict__ bias, float* __restrict__ Out, int do_trunc) {
  __shared__ float xTile[16 * SK_STRIDE];
  const int tid = threadIdx.x;
  const int lane = tid & 31;
  const int wv = tid >> 5;
  const int half = lane >> 4;
  const int l16 = lane & 15;
  const size_t m0 = (size_t)blockIdx.x * 16;

  tile_load_16x512(xTile, X + m0 * D_MODEL, tid);
  __syncthreads();

  v8f acc[4];
  #pragma unroll
  for (int t = 0; t < 4; ++t) acc[t] = zero8();

  const int nbase = wv * 64;
  for (int h0 = 0; h0 < D_MODEL; h0 += 4) {
    const v2f a = *(const v2f*)&xTile[l16 * SK_STRIDE + h0 + 2 * half];
    #pragma unroll
    for (int t = 0; t < 4; ++t) {
      const int n = nbase + t * 16 + l16;
      const v2f bb = *(const v2f*)(W + (size_t)n * D_MODEL + h0 + 2 * half);
      acc[t] = wmma_f32_k4(a, bb, acc[t]);
    }
  }

  #pragma unroll
  for (int t = 0; t < 4; ++t) {
    const int n = nbase + t * 16 + l16;
    const float bv = bias[n];
    #pragma unroll
    for (int r = 0; r < 8; ++r) {
      const size_t row = m0 + r + 8 * half;
      float v = acc[t][r] + bv;
      if (do_trunc) v = truncf(v * (1.0f / SCALE));
      Out[row * D_MODEL + n] = v;
    }
  }
}

// ---------------------------------------------------------------------------
// Online softmax stats over q for each (b,k): masked max & sum-exp.
// Block: (16 k-columns, b) -> 1024 blocks. Each of the 8 waves covers a
// 256-wide q range for the SAME 16 k-columns; partial (m,l) merged via LDS.
// scoreT tile D[m=q, n=k] via WMMA: A = sk rows, B = tq rows (both global,
// L2/L0 resident: the wave's 16 tq rows are reused across its whole q loop).
// ---------------------------------------------------------------------------
__global__ __launch_bounds__(256) void stats_kernel(
    const float* __restrict__ sk, const float* __restrict__ tq,
    const int* __restrict__ mask, float* __restrict__ rowmax,
    float* __restrict__ rowsum) {
  __shared__ float pm[8][16];
  __shared__ float pl[8][16];
  const int tid = threadIdx.x;
  const int lane = tid & 31;
  const int wv = tid >> 5;
  const int half = lane >> 4;
  const int l16 = lane & 15;
  const int b = blockIdx.y;
  const int kb = blockIdx.x * 16;
  const size_t tqRow = ((size_t)b * SEQ + kb + l16) * D_MODEL;

  float m_run = -INFINITY, l_run = 0.0f;

  const int qlo = wv * (SEQ / 8);
  for (int q0 = qlo; q0 < qlo + SEQ / 8; q0 += 16) {
    const size_t skRow = ((size_t)b * SEQ + q0 + l16) * D_MODEL;
    v8f d = zero8();
    for (int h0 = 0; h0 < D_MODEL; h0 += 4) {
      const v2f a = *(const v2f*)(sk + skRow + h0 + 2 * half);
      const v2f bb = *(const v2f*)(tq + tqRow + h0 + 2 * half);
      d = wmma_f32_k4(a, bb, d);
    }

    float vals[8];
    float tmax = -INFINITY;
    #pragma unroll
    for (int r = 0; r < 8; ++r) {
      const bool msk = (mask[b * SEQ + q0 + r + 8 * half] == 0);
      vals[r] = msk ? -INFINITY : d[r];
      tmax = fmaxf(tmax, vals[r]);
    }
    tmax = fmaxf(tmax, __shfl_xor(tmax, 16, 32));  // combine both q-halves
    const float m_new = fmaxf(m_run, tmax);
    if (m_new != -INFINITY) {  // skip if everything so far is masked
      float ps = 0.0f;
      #pragma unroll
      for (int r = 0; r < 8; ++r)
        ps += (vals[r] == -INFINITY) ? 0.0f : __expf(vals[r] - m_new);
      ps += __shfl_xor(ps, 16, 32);
      const float corr = (m_run == -INFINITY) ? 0.0f : __expf(m_run - m_new);
      l_run = l_run * corr + ps;
      m_run = m_new;
    }
  }

  if (lane < 16) {
    pm[wv][l16] = m_run;
    pl[wv][l16] = l_run;
  }
  __syncthreads();
  if (tid < 16) {
    float M = -INFINITY;
    #pragma unroll
    for (int w = 0; w < 8; ++w) M = fmaxf(M, pm[w][tid]);
    float L = 0.0f;
    if (M != -INFINITY) {
      #pragma unroll
      for (int w = 0; w < 8; ++w)
        L += (pm[w][tid] == -INFINITY) ? 0.0f : pl[w][tid] * __expf(pm[w][tid] - M);
    }
    rowmax[(size_t)b * SEQ + kb + tid] = M;
    rowsum[(size_t)b * SEQ + kb + tid] = L;
  }
}

// ---------------------------------------------------------------------------
// out[b,q,n] = 0.5*src + 0.5 * sum_k exp(score[k,q]-m[k])/l[k] * tlt[k,n]
// Block: (q-tile of 16, b). k-loop in 128-wide chunks; wave w recomputes its
// scoreT sub-tile, normalizes, bounces attn through LDS (D-layout->A-layout),
// then all waves chain WMMAs against tlt into 16x512 accumulators.
// ---------------------------------------------------------------------------
__global__ __launch_bounds__(256) void attnout_kernel(
    const float* __restrict__ sk, const float* __restrict__ tq,
    const float* __restrict__ tlt, const float* __restrict__ src,
    const int* __restrict__ mask, const float* __restrict__ rowmax,
    const float* __restrict__ rowsum, float* __restrict__ out) {
  __shared__ float skTile[16 * SK_STRIDE];
  __shared__ float attnTile[16 * ATT_STRIDE];
  __shared__ int mTile[16];
  const int tid = threadIdx.x;
  const int lane = tid & 31;
  const int wv = tid >> 5;
  const int half = lane >> 4;
  const int l16 = lane & 15;
  const int b = blockIdx.y;
  const int q0 = blockIdx.x * 16;

  tile_load_16x512(skTile, sk + ((size_t)b * SEQ + q0) * D_MODEL, tid);
  if (tid < 16) mTile[tid] = mask[b * SEQ + q0 + tid];
  __syncthreads();

  v8f acc[4];
  #pragma unroll
  for (int t = 0; t < 4; ++t) acc[t] = zero8();
  const int nbase = wv * 64;

  for (int kc = 0; kc < SEQ; kc += 128) {
    const int kb = kc + wv * 16;
    const size_t krow = (size_t)b * SEQ + kb + l16;
    const size_t tqRow = krow * D_MODEL;

    v8f d = zero8();
    for (int h0 = 0; h0 < D_MODEL; h0 += 4) {
      const v2f a = *(const v2f*)&skTile[l16 * SK_STRIDE + h0 + 2 * half];
      const v2f bb = *(const v2f*)(tq + tqRow + h0 + 2 * half);
      d = wmma_f32_k4(a, bb, d);
    }

    const float mv = rowmax[krow];
    const float lv = rowsum[krow];
    const float inv = (lv > 0.0f) ? 1.0f / lv : 0.0f;
    #pragma unroll
    for (int r = 0; r < 8; ++r) {
      const bool msk = (mTile[r + 8 * half] == 0);
      const float a = (!msk && inv > 0.0f) ? __expf(d[r] - mv) * inv : 0.0f;
      attnTile[(r + 8 * half) * ATT_STRIDE + wv * 16 + l16] = a;
    }
    __syncthreads();  // attn tile visible to all waves

    for (int kk = 0; kk < 128; kk += 4) {
      const v2f a = *(const v2f*)&attnTile[l16 * ATT_STRIDE + kk + 2 * half];
      const size_t trow = ((size_t)b * SEQ + kc + kk + 2 * half) * D_MODEL;
      #pragma unroll
      for (int t = 0; t < 4; ++t) {
        const int n = nbase + t * 16 + l16;
        v2f bb;
        bb.x = tlt[trow + n];
        bb.y = tlt[trow + D_MODEL + n];
        acc[t] = wmma_f32_k4(a, bb, acc[t]);
      }
    }
    __syncthreads();  // done reading attn tile before next chunk overwrites
  }

  #pragma unroll
  for (int t = 0; t < 4; ++t) {
    const int n = nbase + t * 16 + l16;
    #pragma unroll
    for (int r = 0; r < 8; ++r) {
      const size_t row = (size_t)b * SEQ + q0 + r + 8 * half;
      out[row * D_MODEL + n] = 0.5f * src[row * D_MODEL + n] + 0.5f * acc[t][r];
    }
  }
}

extern "C" void kernel_launch(void* const* d_in, const int* in_sizes, int n_in,
                              void* d_out, int out_size, void* d_ws,
                              size_t ws_size, hipStream_t stream) {
  const float* src = (const float*)d_in[0];   // [B,Q,D]
  const float* tlt = (const float*)d_in[1];   // [B,K,D]
  const int* mask = (const int*)d_in[2];      // [B,Q]
  const float* Wq = (const float*)d_in[3];    // [D,D]
  const float* bq = (const float*)d_in[4];    // [D]
  const float* Wk = (const float*)d_in[5];    // [D,D]
  const float* bk = (const float*)d_in[6];    // [D]
  float* out = (float*)d_out;                 // [B,Q,D]

  float* tq = (float*)d_ws;
  float* sk = tq + (size_t)BATCH * SEQ * D_MODEL;
  float* rowmax = sk + (size_t)BATCH * SEQ * D_MODEL;
  float* rowsum = rowmax + (size_t)BATCH * SEQ;

  const dim3 blk(256);
  proj_kernel<<<dim3(BATCH * SEQ / 16), blk, 0, stream>>>(tlt, Wq, bq, tq, 1);
  proj_kernel<<<dim3(BATCH * SEQ / 16), blk, 0, stream>>>(src, Wk, bk, sk, 0);
  stats_kernel<<<dim3(SEQ / 16, BATCH), blk, 0, stream>>>(sk, tq, mask,
                                                          rowmax, rowsum);
  attnout_kernel<<<dim3(SEQ / 16, BATCH), blk, 0, stream>>>(
      sk, tq, tlt, src, mask, rowmax, rowsum, out);
}